// Sparsemax_77661598646585
// MI455X (gfx1250) — compile-verified
//
#include <hip/hip_runtime.h>

// Sparsemax over last axis, shape (2048*2048 rows) x (k=32), fp32.
//
// Roofline: 1 GiB total traffic / 23.3 TB/s => ~46 us floor. Strategy:
//  - one ROW per LANE (row k=32 lives in 32 VGPRs), so each VALU op covers
//    32 rows: ~18 wave-instructions per row total.
//  - gfx1250 async global->LDS engine stages each wave's 32x32 tile with
//    fully coalesced 512B global requests; padded LDS layout (stride 36
//    dwords) gives 16B-aligned rows for ds_*_b128 with minimal conflicts.
//  - tau via simplex-projection identity tau = max_j (S_j - 1)/j on the
//    descending-sorted row (Batcher odd-even mergesort, 191 comparators,
//    branch-free v_min/v_max pairs).

#define K 32
#define ROWS_PER_BLOCK 256   // 8 wave32s per block
#define STRIDE 36            // dwords per row in LDS: 144B (16B-aligned rows)

__device__ __forceinline__ void cswap_desc(float& a, float& b) {
  const float hi = fmaxf(a, b);
  const float lo = fminf(a, b);
  a = hi;
  b = lo;
}

__global__ __launch_bounds__(ROWS_PER_BLOCK) void sparsemax32_kernel(
    const float* __restrict__ x, float* __restrict__ out, unsigned nrows) {
  __shared__ float sm[ROWS_PER_BLOCK * STRIDE];  // 36,864 B

  const unsigned tid = threadIdx.x;
  const unsigned wv = tid >> 5;
  const unsigned lane = tid & 31u;
  const unsigned wave_row0 = blockIdx.x * ROWS_PER_BLOCK + wv * 32u;
  if (wave_row0 >= nrows) return;  // whole-wave exit (nrows % 32 == 0)

  // ---- Stage 1: async global -> LDS copy of this wave's 32x32 f32 tile ----
  // Flat 16B chunk c = q*32 + lane : global byte  = tile_base + c*16
  //                                  row          = c / 8   (4 rows per q)
  //                                  lds byte     = row*144 + (c%8)*16
  const unsigned g0 = wave_row0 * (K * 4u) + lane * 16u;  // byte offset in x
  const unsigned lds_wave =
      (unsigned)(unsigned long long)(&sm[0]) + wv * 32u * (STRIDE * 4u);
  const unsigned lds0 =
      lds_wave + (lane >> 3) * (STRIDE * 4u) + (lane & 7u) * 16u;

#pragma unroll
  for (unsigned q = 0; q < 8; ++q) {
    asm volatile("global_load_async_to_lds_b128 %0, %1, %2"
                 :
                 : "v"(lds0 + q * (4u * STRIDE * 4u)),  // +4 rows = 576 B
                   "v"(g0 + q * 512u),
                   "s"(x)
                 : "memory");
  }
  asm volatile("s_wait_asynccnt 0x0" ::: "memory");

  // ---- Stage 2: per-lane row load (contiguous, 16B aligned) ----
  float* rowp = (float*)&sm[0] + (wv * 32u + lane) * STRIDE;
  float v[K];
#pragma unroll
  for (int q = 0; q < 8; ++q) {
    const float4 t = *reinterpret_cast<const float4*>(rowp + 4 * q);
    v[4 * q + 0] = t.x;
    v[4 * q + 1] = t.y;
    v[4 * q + 2] = t.z;
    v[4 * q + 3] = t.w;
  }

  // ---- Stage 3: Batcher odd-even mergesort, descending (191 comparators) --
#pragma unroll
  for (int p = 1; p < K; p <<= 1) {
#pragma unroll
    for (int k = p; k >= 1; k >>= 1) {
#pragma unroll
      for (int j = k % p; j + k < K; j += 2 * k) {
#pragma unroll
        for (int i = 0; i < k; ++i) {
          const int a = i + j, b = i + j + k;
          if (b < K && (a / (2 * p) == b / (2 * p))) cswap_desc(v[a], v[b]);
        }
      }
    }
  }

  // ---- Stage 4: tau = max_j (S_j - 1)/j  (S_j = prefix sum, descending) ---
  float S = v[0];
  float tau = v[0] - 1.0f;
#pragma unroll
  for (int j = 1; j < K; ++j) {
    S += v[j];
    const float c = 1.0f / (float)(j + 1);  // compile-time constant
    tau = fmaxf(tau, __builtin_fmaf(S, c, -c));  // (S - 1)/(j+1)
  }

  // ---- Stage 5: out = max(x - tau, 0), staged back through LDS ----
#pragma unroll
  for (int q = 0; q < 8; ++q) {
    float4 t = *reinterpret_cast<const float4*>(rowp + 4 * q);
    t.x = fmaxf(t.x - tau, 0.0f);
    t.y = fmaxf(t.y - tau, 0.0f);
    t.z = fmaxf(t.z - tau, 0.0f);
    t.w = fmaxf(t.w - tau, 0.0f);
    *reinterpret_cast<float4*>(rowp + 4 * q) = t;
  }
  asm volatile("s_wait_dscnt 0x0" ::: "memory");

  // ---- Stage 6: async LDS -> global, coalesced 512B stores ----
#pragma unroll
  for (unsigned q = 0; q < 8; ++q) {
    asm volatile("global_store_async_from_lds_b128 %0, %1, %2"
                 :
                 : "v"(g0 + q * 512u),
                   "v"(lds0 + q * (4u * STRIDE * 4u)),
                   "s"(out)
                 : "memory");
  }
  // s_endpgm performs an implicit wait-idle, draining ASYNCcnt.
}

extern "C" void kernel_launch(void* const* d_in, const int* in_sizes, int n_in,
                              void* d_out, int out_size, void* d_ws,
                              size_t ws_size, hipStream_t stream) {
  const float* x = (const float*)d_in[0];
  float* out = (float*)d_out;
  const unsigned nrows = (unsigned)(in_sizes[0] / K);  // 4,194,304
  const unsigned blocks = (nrows + ROWS_PER_BLOCK - 1) / ROWS_PER_BLOCK;
  sparsemax32_kernel<<<blocks, ROWS_PER_BLOCK, 0, stream>>>(x, out, nrows);
}